// SimpleGiNetwork_28003186770215
// MI455X (gfx1250) — compile-verified
//
#include <hip/hip_runtime.h>
#include <hip/hip_bf16.h>

// ============================================================================
// SimpleGiNetwork on MI455X (gfx1250)
//
// Key identity: softmax over an [E,1] axis == 1.0 exactly, and both the gather
// and scatter of _message use ei[0]. So each message layer reduces to the
// per-node dense op   h' = relu(deg0[v] * (W h[v] + b)),
// with deg0 = histogram(ei[0]).  The edge-sized tensors (attr, fn(x[ei1]),
// fe, fa) are mathematically dead.  The remaining work is ~1 GFLOP of tiny
// per-node GEMMs + ~80 MB of traffic -> bandwidth bound; keep f32 and use
// V_WMMA_F32_16X16X4_F32 for the dense math.
// ============================================================================

#define N_NODES      200000
#define N_EDGES      4000000
#define N_INT_EDGES  2000000
#define N_GRAPHS     256
#define N_TILES      (N_NODES / 16)   // 12500, exact (no tail tile)
#define WAVES_PER_WG 8
#define SLAB_STRIDE  20               // padded row stride (words) -> no LDS bank conflicts

typedef __attribute__((ext_vector_type(2))) float v2f;
typedef __attribute__((ext_vector_type(8))) float v8f;

// ---------------------------------------------------------------------------
__global__ void zero_ws_kernel(unsigned* __restrict__ p, int n) {
  for (int i = blockIdx.x * blockDim.x + threadIdx.x; i < n;
       i += gridDim.x * blockDim.x)
    p[i] = 0u;
}

// degree / count histogram: global_atomic_add_u32
__global__ void hist_kernel(const int* __restrict__ idx, int n,
                            unsigned* __restrict__ out) {
  for (int i = blockIdx.x * blockDim.x + threadIdx.x; i < n;
       i += gridDim.x * blockDim.x)
    atomicAdd(&out[idx[i]], 1u);
}

// ---------------------------------------------------------------------------
// One wave32 owns a 16-node tile and runs the full 3-layer chain:
//   L1: [16x32]x[32x16] (8 WMMA k-steps), L2: [16x16]x[16x16] (4),
//   L3: [16x16]x[16x32] (2 N-tiles x 4).
// A-matrix f32 16x4 layout: lane<16 holds {K=0,1}, lane>=16 holds {K=2,3},
// row M = lane&15 -> a contiguous float2 per lane per k-chunk.
// B mirrors A with N = lane&15 (B[k][n] = W[n][k], W is row-major [out,in]).
// C/D layout: VGPR r, lane l -> (M = r + 8*(l>>4), N = l&15).
// ---------------------------------------------------------------------------
__global__ __launch_bounds__(256) void gnn_chain_kernel(
    const float* __restrict__ x,          // [N_NODES, 32]
    const unsigned* __restrict__ deg,     // [N_NODES]
    const int* __restrict__ batch,        // [N_NODES] (sorted)
    const float* __restrict__ W1, const float* __restrict__ b1,  // [16,32],[16]
    const float* __restrict__ W2, const float* __restrict__ b2,  // [16,16],[16]
    const float* __restrict__ W3, const float* __restrict__ b3,  // [32,16],[32]
    float* __restrict__ acc)              // [N_GRAPHS, 32] f32 accumulators
{
  __shared__ float lds[WAVES_PER_WG * 16 * SLAB_STRIDE];

  const int lane = threadIdx.x & 31;
  const int wave = threadIdx.x >> 5;
  const int tile = blockIdx.x * WAVES_PER_WG + wave;
  if (tile >= N_TILES) return;            // wave-uniform exit: EXEC stays all-1s

  const int hi = lane >> 4;               // 0 | 1
  const int lo = lane & 15;
  const int nb = tile * 16;               // first node of this tile
  float* slab = &lds[wave * 16 * SLAB_STRIDE];

  // Per-row degree scale + graph id (row M = r + 8*hi); loads broadcast in-wave.
  float dscale[8];
  int   gid[8];
#pragma unroll
  for (int r = 0; r < 8; ++r) {
    const int node = nb + r + 8 * hi;
    dscale[r] = (float)deg[node];
    gid[r]    = batch[node];
  }

  // ---------------- Layer 1: 32 -> 16 --------------------------------------
  v8f c1;
  {
    const float bias = b1[lo];
#pragma unroll
    for (int r = 0; r < 8; ++r) c1[r] = bias;   // C starts at broadcast bias
  }
#pragma unroll
  for (int kc = 0; kc < 8; ++kc) {
    const int k = kc * 4 + 2 * hi;
    const v2f a = *(const v2f*)&x[(nb + lo) * 32 + k];
    const v2f b = *(const v2f*)&W1[lo * 32 + k];
    c1 = __builtin_amdgcn_wmma_f32_16x16x4_f32(false, a, false, b,
                                               (short)0, c1, false, false);
  }
  // epilogue: relu(deg * (Wx + b)) -> LDS [node][feat] for next layer's A
#pragma unroll
  for (int r = 0; r < 8; ++r)
    slab[(r + 8 * hi) * SLAB_STRIDE + lo] = fmaxf(dscale[r] * c1[r], 0.0f);
  __builtin_amdgcn_wave_barrier();

  // ---------------- Layer 2: 16 -> 16 --------------------------------------
  v8f c2;
  {
    const float bias = b2[lo];
#pragma unroll
    for (int r = 0; r < 8; ++r) c2[r] = bias;
  }
#pragma unroll
  for (int kc = 0; kc < 4; ++kc) {
    const int k = kc * 4 + 2 * hi;
    const v2f a = *(const v2f*)&slab[lo * SLAB_STRIDE + k];
    const v2f b = *(const v2f*)&W2[lo * 16 + k];
    c2 = __builtin_amdgcn_wmma_f32_16x16x4_f32(false, a, false, b,
                                               (short)0, c2, false, false);
  }
  __builtin_amdgcn_wave_barrier();        // all slab reads precede the rewrite
#pragma unroll
  for (int r = 0; r < 8; ++r)
    slab[(r + 8 * hi) * SLAB_STRIDE + lo] = fmaxf(dscale[r] * c2[r], 0.0f);
  __builtin_amdgcn_wave_barrier();

  // ---------------- Layer 3: 16 -> 32 (two N-tiles) -------------------------
  v8f c3a, c3b;
  {
    const float biasA = b3[lo];
    const float biasB = b3[16 + lo];
#pragma unroll
    for (int r = 0; r < 8; ++r) { c3a[r] = biasA; c3b[r] = biasB; }
  }
#pragma unroll
  for (int kc = 0; kc < 4; ++kc) {
    const int k = kc * 4 + 2 * hi;
    const v2f a  = *(const v2f*)&slab[lo * SLAB_STRIDE + k];
    const v2f bA = *(const v2f*)&W3[lo * 16 + k];
    const v2f bB = *(const v2f*)&W3[(16 + lo) * 16 + k];
    c3a = __builtin_amdgcn_wmma_f32_16x16x4_f32(false, a, false, bA,
                                                (short)0, c3a, false, false);
    c3b = __builtin_amdgcn_wmma_f32_16x16x4_f32(false, a, false, bB,
                                                (short)0, c3b, false, false);
  }

  // ---- per-graph accumulation (segment-sum numerator of the mean) ----------
  // batch is sorted -> a tile's 8 rows almost always share one graph: fold the
  // 8 row-contributions per lane into a single global_atomic_add_f32.
  bool uni = true;
#pragma unroll
  for (int r = 1; r < 8; ++r) uni = uni && (gid[r] == gid[0]);

  if (uni) {
    float sA = 0.0f, sB = 0.0f;
#pragma unroll
    for (int r = 0; r < 8; ++r) {
      sA += fmaxf(dscale[r] * c3a[r], 0.0f);
      sB += fmaxf(dscale[r] * c3b[r], 0.0f);
    }
    atomicAdd(&acc[gid[0] * 32 + lo],      sA);
    atomicAdd(&acc[gid[0] * 32 + 16 + lo], sB);
  } else {
#pragma unroll
    for (int r = 0; r < 8; ++r) {
      atomicAdd(&acc[gid[r] * 32 + lo],      fmaxf(dscale[r] * c3a[r], 0.0f));
      atomicAdd(&acc[gid[r] * 32 + 16 + lo], fmaxf(dscale[r] * c3b[r], 0.0f));
    }
  }
}

// ---------------------------------------------------------------------------
// Head: per-graph mean -> concat(64) -> fc1(64->128)+relu -> fc2(128->2).
// 256 blocks x 128 threads; ~4 MFLOP total (negligible), plain VALU.
// ---------------------------------------------------------------------------
__global__ __launch_bounds__(128) void final_head_kernel(
    const float* __restrict__ accH, const float* __restrict__ accG,
    const unsigned* __restrict__ cnt,
    const float* __restrict__ fc1w, const float* __restrict__ fc1b,  // [128,64],[128]
    const float* __restrict__ fc2w, const float* __restrict__ fc2b,  // [2,128],[2]
    float* __restrict__ out)                                         // [256,2]
{
  __shared__ float p[64];
  __shared__ float o1[128];
  const int g = blockIdx.x;
  const int t = threadIdx.x;

  const float inv = 1.0f / fmaxf((float)cnt[g], 1.0f);
  if (t < 32) {
    p[t]      = accH[g * 32 + t] * inv;
    p[32 + t] = accG[g * 32 + t] * inv;
  }
  __syncthreads();

  float s = fc1b[t];
#pragma unroll 8
  for (int k = 0; k < 64; ++k) s += p[k] * fc1w[t * 64 + k];
  o1[t] = fmaxf(s, 0.0f);
  __syncthreads();

  if (t < 2) {
    float s2 = fc2b[t];
#pragma unroll 8
    for (int k = 0; k < 128; ++k) s2 += o1[k] * fc2w[t * 128 + k];
    out[g * 2 + t] = s2;
  }
}

// ===========================================================================
extern "C" void kernel_launch(void* const* d_in, const int* in_sizes, int n_in,
                              void* d_out, int out_size, void* d_ws, size_t ws_size,
                              hipStream_t stream) {
  (void)out_size; (void)ws_size;

  // ---- fixed top-level positions (setup_inputs insertion order; params is
  //      40 leaves at indices 3..42 under either pytree convention) ----------
  const float* x          = (const float*)d_in[0];
  const int*   edge_idx0  = (const int*)d_in[43];   // row 0 of [2, N_EDGES]
  const int*   int_idx0   = (const int*)d_in[44];   // row 0 of [2, N_INT_EDGES]
  const int*   batch      = (const int*)d_in[45];

  // ---- params pytree order detection --------------------------------------
  // sorted-keys (jax tree_flatten): edge@3..20, fc1@21/22, fc2@23/24, internal@25..42
  //   per msg: fa.b, fa.w, fe.b, fe.w, fn.b, fn.w   (fn.b=+4, fn.w=+5)
  // insertion order: internal@3..20, edge@21..38, fc1@39/40, fc2@41/42
  //   per msg: fn.w, fn.b, fe.w, fe.b, fa.w, fa.b   (fn.w=+0, fn.b=+1)
  // Discriminator: in_sizes[21] == 128 (fc1.b) only in the sorted layout.
  const bool sorted_keys = (n_in > 21) && (in_sizes[21] == 128);

  const float *iW[3], *ib_[3], *eW[3], *eb_[3];
  const float *fc1w, *fc1b, *fc2w, *fc2b;
  if (sorted_keys) {
    for (int l = 0; l < 3; ++l) {
      eb_[l] = (const float*)d_in[3 + 6 * l + 4];
      eW[l]  = (const float*)d_in[3 + 6 * l + 5];
      ib_[l] = (const float*)d_in[25 + 6 * l + 4];
      iW[l]  = (const float*)d_in[25 + 6 * l + 5];
    }
    fc1b = (const float*)d_in[21]; fc1w = (const float*)d_in[22];
    fc2b = (const float*)d_in[23]; fc2w = (const float*)d_in[24];
  } else {
    for (int l = 0; l < 3; ++l) {
      iW[l]  = (const float*)d_in[3 + 6 * l + 0];
      ib_[l] = (const float*)d_in[3 + 6 * l + 1];
      eW[l]  = (const float*)d_in[21 + 6 * l + 0];
      eb_[l] = (const float*)d_in[21 + 6 * l + 1];
    }
    fc1w = (const float*)d_in[39]; fc1b = (const float*)d_in[40];
    fc2w = (const float*)d_in[41]; fc2b = (const float*)d_in[42];
  }

  // ---- workspace layout (u32/f32 words) -----------------------------------
  unsigned* degI = (unsigned*)d_ws;                  // [N_NODES]
  unsigned* degE = degI + N_NODES;                   // [N_NODES]
  unsigned* cnt  = degE + N_NODES;                   // [N_GRAPHS] (padded to 256)
  float*    accH = (float*)(cnt + 256);              // [N_GRAPHS*32]
  float*    accG = accH + N_GRAPHS * 32;             // [N_GRAPHS*32]
  const int ws_words = 2 * N_NODES + 256 + 2 * N_GRAPHS * 32;

  // 1) zero accumulators (required every replay)
  zero_ws_kernel<<<1024, 256, 0, stream>>>((unsigned*)d_ws, ws_words);

  // 2) degree + graph-count histograms
  hist_kernel<<<2048, 256, 0, stream>>>(edge_idx0, N_EDGES,     degE);
  hist_kernel<<<2048, 256, 0, stream>>>(int_idx0,  N_INT_EDGES, degI);
  hist_kernel<<<512,  256, 0, stream>>>(batch,     N_NODES,     cnt);

  // 3) the two per-node WMMA chains (internal -> h, edge -> g)
  const int grid = (N_TILES + WAVES_PER_WG - 1) / WAVES_PER_WG;  // 1563
  gnn_chain_kernel<<<grid, 256, 0, stream>>>(
      x, degI, batch, iW[0], ib_[0], iW[1], ib_[1], iW[2], ib_[2], accH);
  gnn_chain_kernel<<<grid, 256, 0, stream>>>(
      x, degE, batch, eW[0], eb_[0], eW[1], eb_[1], eW[2], eb_[2], accG);

  // 4) head
  final_head_kernel<<<N_GRAPHS, 128, 0, stream>>>(
      accH, accG, cnt, fc1w, fc1b, fc2w, fc2b, (float*)d_out);
}